// RGAT_Net_71975061946537
// MI455X (gfx1250) — compile-verified
//
#include <hip/hip_runtime.h>
#include <hip/hip_bf16.h>

typedef __attribute__((ext_vector_type(2))) float v2f;
typedef __attribute__((ext_vector_type(8))) float v8f;

#define N_NODES 100000
#define N_EDGES 1600000
#define N_REL   8

// ---------------------------------------------------------------------------
// Zero-fill scratch (d_ws / d_out are NOT re-poisoned between graph replays)
// ---------------------------------------------------------------------------
__global__ void zero_kernel(float* __restrict__ p, long long n) {
    long long i = (long long)blockIdx.x * blockDim.x + threadIdx.x;
    if (i < n) p[i] = 0.0f;
}

// ---------------------------------------------------------------------------
// Fold Q/K into W:  WQ[r][i] = sum_o W[r][i][o] * Q[o]   (tiny)
// ---------------------------------------------------------------------------
__global__ void wqk_kernel(const float* __restrict__ W, const float* __restrict__ Q,
                           const float* __restrict__ K, float* __restrict__ WQ,
                           float* __restrict__ WK, int Cin, int Cout) {
    int idx = blockIdx.x * blockDim.x + threadIdx.x;   // r*Cin + i
    if (idx >= N_REL * Cin) return;
    const float* w = W + (size_t)idx * Cout;
    float sq = 0.f, sk = 0.f;
    for (int o = 0; o < Cout; ++o) { sq += w[o] * Q[o]; sk += w[o] * K[o]; }
    WQ[idx] = sq; WK[idx] = sk;
}

// ---------------------------------------------------------------------------
// Per-node attention logits for all relations: q[r,n] = x[n,:] . WQ[r,:]
// ---------------------------------------------------------------------------
template<int Cin>
__global__ void qk_kernel(const float* __restrict__ X, const float* __restrict__ WQ,
                          const float* __restrict__ WK, float* __restrict__ q,
                          float* __restrict__ k, int N) {
    __shared__ float sWQ[N_REL * Cin];
    __shared__ float sWK[N_REL * Cin];
    for (int t = threadIdx.x; t < N_REL * Cin; t += blockDim.x) {
        sWQ[t] = WQ[t]; sWK[t] = WK[t];
    }
    __syncthreads();
    int n = blockIdx.x * blockDim.x + threadIdx.x;
    if (n >= N) return;
    float xr[Cin];
    const float* xp = X + (size_t)n * Cin;
    #pragma unroll
    for (int i = 0; i < Cin; ++i) xr[i] = xp[i];
    #pragma unroll
    for (int r = 0; r < N_REL; ++r) {
        float sq = 0.f, sk = 0.f;
        #pragma unroll
        for (int i = 0; i < Cin; ++i) {
            sq += xr[i] * sWQ[r * Cin + i];
            sk += xr[i] * sWK[r * Cin + i];
        }
        q[(size_t)r * N + n] = sq;
        k[(size_t)r * N + n] = sk;
    }
}

// ---------------------------------------------------------------------------
// Relation-typed GEMM via V_WMMA_F32_16X16X4_F32.
// One wave computes one 16x16 tile of XW[r] = X @ W[r], K-loop step 4.
// A (16x4 f32): lanes 0-15 hold M=lane, VGPR0/1 = K0/K1; lanes 16-31: K2/K3.
// B (4x16 f32): mirrored; C/D: 8 VGPRs, lanes 0-15 rows M=0..7, 16-31 M=8..15.
// ---------------------------------------------------------------------------
template<int Cin, int Cout>
__global__ void wmma_gemm_rel(const float* __restrict__ X, const float* __restrict__ W,
                              float* __restrict__ XW, int N) {
    const int wavesPerBlock = blockDim.x >> 5;
    int wave = blockIdx.x * wavesPerBlock + (threadIdx.x >> 5);
    int lane = threadIdx.x & 31;
    const int tn = Cout >> 4;          // tiles along output channels
    const int tm = N >> 4;             // tiles along nodes (100000/16 = 6250 exact)
    int total = N_REL * tm * tn;
    if (wave >= total) return;         // whole-wave uniform; live waves have EXEC all-1s
    int r   = wave / (tm * tn);
    int rem = wave % (tm * tn);
    int mt  = rem / tn;
    int nt  = rem % tn;
    int m0 = mt << 4, n0 = nt << 4;
    int half = lane >> 4, lr = lane & 15;

    const float* Xr = X + (size_t)(m0 + lr) * Cin;
    const float* Wr = W + (size_t)r * Cin * Cout;

    v8f c = {};
    #pragma unroll
    for (int kk = 0; kk < Cin; kk += 4) {
        int ka = kk + (half << 1);
        v2f a, b;
        a.x = Xr[ka];
        a.y = Xr[ka + 1];
        b.x = Wr[(size_t)ka       * Cout + n0 + lr];
        b.y = Wr[(size_t)(ka + 1) * Cout + n0 + lr];
        c = __builtin_amdgcn_wmma_f32_16x16x4_f32(
                /*neg_a=*/false, a, /*neg_b=*/false, b,
                /*c_mod=*/(short)0, c, /*reuse_a=*/false, /*reuse_b=*/false);
    }

    float* out = XW + ((size_t)r * N + m0) * Cout + n0 + lr;
    #pragma unroll
    for (int i = 0; i < 8; ++i)
        out[(size_t)(i + 8 * half) * Cout] = c[i];
}

// ---------------------------------------------------------------------------
// Order-preserving uint encoding of float for atomic max
// ---------------------------------------------------------------------------
__device__ __forceinline__ unsigned encf(float f) {
    unsigned u = __float_as_uint(f);
    return (u >> 31) ? ~u : (u | 0x80000000u);
}
__device__ __forceinline__ float decf(unsigned v) {
    return (v >> 31) ? __uint_as_float(v & 0x7fffffffu) : __uint_as_float(~v);
}

// Pass A: segment-max of leaky-relu logits per dst node
__global__ void edge_max_kernel(const int* __restrict__ src, const int* __restrict__ dst,
                                const int* __restrict__ et, const float* __restrict__ q,
                                const float* __restrict__ k, unsigned* __restrict__ amaxU,
                                int E, int N) {
    int e = blockIdx.x * blockDim.x + threadIdx.x;
    if (e >= E) return;
    int s = src[e], d = dst[e], r = et[e];
    float a = q[(size_t)r * N + d] + k[(size_t)r * N + s];
    a = (a > 0.f) ? a : 0.2f * a;
    atomicMax(amaxU + d, encf(a));
}

// Pass B: w = exp(alpha - amax[dst]); denom[dst] += w; acc[dst,:] += w * xw[et,src,:]
// One wave per edge, lanes strided over channels.
template<int C>
__global__ void edge_accum_kernel(const int* __restrict__ src, const int* __restrict__ dst,
                                  const int* __restrict__ et, const float* __restrict__ q,
                                  const float* __restrict__ k, const unsigned* __restrict__ amaxU,
                                  const float* __restrict__ xw, float* __restrict__ denom,
                                  float* __restrict__ acc, int E, int N) {
    int wave = blockIdx.x * (blockDim.x >> 5) + (threadIdx.x >> 5);
    int lane = threadIdx.x & 31;
    if (wave >= E) return;
    int s = src[wave], d = dst[wave], r = et[wave];
    float a = q[(size_t)r * N + d] + k[(size_t)r * N + s];
    a = (a > 0.f) ? a : 0.2f * a;
    float w = __expf(a - decf(amaxU[d]));
    if (lane == 0) atomicAdd(denom + d, w);
    const float* row = xw + ((size_t)r * N + s) * C;
    float* o = acc + (size_t)d * C;
    #pragma unroll
    for (int c = lane; c < C; c += 32) atomicAdd(o + c, w * row[c]);
}

// out[n,c] = acc[n,c] / (denom[n] + 1e-16) + b[c]; optional ReLU
template<bool RELU>
__global__ void finalize_kernel(const float* __restrict__ denom, const float* __restrict__ b,
                                float* __restrict__ acc, int N, int C) {
    long long idx = (long long)blockIdx.x * blockDim.x + threadIdx.x;
    if (idx >= (long long)N * C) return;
    int n = (int)(idx / C), c = (int)(idx % C);
    float v = acc[idx] / (denom[n] + 1e-16f) + b[c];
    if (RELU) v = v > 0.f ? v : 0.f;
    acc[idx] = v;
}

// ---------------------------------------------------------------------------
extern "C" void kernel_launch(void* const* d_in, const int* in_sizes, int n_in,
                              void* d_out, int out_size, void* d_ws, size_t ws_size,
                              hipStream_t stream) {
    const float* x  = (const float*)d_in[0];
    const int*   ei = (const int*)  d_in[1];
    const int*   et = (const int*)  d_in[2];
    const float* W0 = (const float*)d_in[3];
    const float* Q0 = (const float*)d_in[4];
    const float* K0 = (const float*)d_in[5];
    const float* b0 = (const float*)d_in[6];
    const float* W1 = (const float*)d_in[7];
    const float* Q1 = (const float*)d_in[8];
    const float* K1 = (const float*)d_in[9];
    const float* b1 = (const float*)d_in[10];
    float* out = (float*)d_out;

    const int N = N_NODES, E = N_EDGES, R = N_REL, C0 = 64, C1 = 32;
    const int* src = ei;
    const int* dst = ei + E;

    // workspace carve-up (256B aligned)
    char* ws = (char*)d_ws;
    size_t off = 0;
    auto carve = [&](size_t bytes) {
        void* p = ws + off;
        off += (bytes + 255) & ~(size_t)255;
        return p;
    };
    float*    xw    = (float*)   carve((size_t)R * N * C0 * 4);  // reused by layer 1
    float*    q     = (float*)   carve((size_t)R * N * 4);
    float*    k     = (float*)   carve((size_t)R * N * 4);
    unsigned* amaxU = (unsigned*)carve((size_t)N * 4);
    float*    denom = (float*)   carve((size_t)N * 4);
    float*    h     = (float*)   carve((size_t)N * C0 * 4);
    float*    WQ    = (float*)   carve((size_t)R * C0 * 4);
    float*    WK    = (float*)   carve((size_t)R * C0 * 4);

    const int BT = 256;

    // ======================= layer 0 (64 -> 64) =======================
    wqk_kernel<<<(R * C0 + BT - 1) / BT, BT, 0, stream>>>(W0, Q0, K0, WQ, WK, C0, C0);
    qk_kernel<64><<<(N + BT - 1) / BT, BT, 0, stream>>>(x, WQ, WK, q, k, N);
    // 8 * 6250 * 4 = 200000 waves, 8 waves/block -> 25000 blocks (exact)
    wmma_gemm_rel<64, 64><<<(R * (N / 16) * (C0 / 16)) / 8, BT, 0, stream>>>(x, W0, xw, N);

    zero_kernel<<<(N + BT - 1) / BT, BT, 0, stream>>>((float*)amaxU, N);
    zero_kernel<<<(N + BT - 1) / BT, BT, 0, stream>>>(denom, N);
    zero_kernel<<<((long long)N * C0 + BT - 1) / BT, BT, 0, stream>>>(h, (long long)N * C0);

    edge_max_kernel<<<(E + BT - 1) / BT, BT, 0, stream>>>(src, dst, et, q, k, amaxU, E, N);
    edge_accum_kernel<64><<<E / 8, BT, 0, stream>>>(src, dst, et, q, k, amaxU, xw, denom, h, E, N);
    finalize_kernel<true><<<((long long)N * C0 + BT - 1) / BT, BT, 0, stream>>>(denom, b0, h, N, C0);

    // ======================= layer 1 (64 -> 32) =======================
    wqk_kernel<<<(R * C0 + BT - 1) / BT, BT, 0, stream>>>(W1, Q1, K1, WQ, WK, C0, C1);
    qk_kernel<64><<<(N + BT - 1) / BT, BT, 0, stream>>>(h, WQ, WK, q, k, N);
    // 8 * 6250 * 2 = 100000 waves -> 12500 blocks (exact)
    wmma_gemm_rel<64, 32><<<(R * (N / 16) * (C1 / 16)) / 8, BT, 0, stream>>>(h, W1, xw, N);

    zero_kernel<<<(N + BT - 1) / BT, BT, 0, stream>>>((float*)amaxU, N);
    zero_kernel<<<(N + BT - 1) / BT, BT, 0, stream>>>(denom, N);
    zero_kernel<<<((long long)N * C1 + BT - 1) / BT, BT, 0, stream>>>(out, (long long)N * C1);

    edge_max_kernel<<<(E + BT - 1) / BT, BT, 0, stream>>>(src, dst, et, q, k, amaxU, E, N);
    edge_accum_kernel<32><<<E / 8, BT, 0, stream>>>(src, dst, et, q, k, amaxU, xw, denom, out, E, N);
    finalize_kernel<false><<<((long long)N * C1 + BT - 1) / BT, BT, 0, stream>>>(denom, b1, out, N, C1);
}